// OutputBlock_27994596835750
// MI455X (gfx1250) — compile-verified
//
#include <hip/hip_runtime.h>
#include <hip/hip_bf16.h>
#include <math.h>

#define HIDDEN 128
#define NUM_RADIAL 16

typedef __attribute__((ext_vector_type(2))) float v2f;
typedef __attribute__((ext_vector_type(8))) float v8f;

// D = A(16x4, f32) * B(4x16, f32) + C(16x16, f32)
// 8-arg form: (neg_a, A, neg_b, B, c_mod, C, reuse_a, reuse_b)
__device__ __forceinline__ v8f wmma_f32_16x16x4(v2f a, v2f b, v8f c) {
    return __builtin_amdgcn_wmma_f32_16x16x4_f32(
        /*neg_a=*/false, a, /*neg_b=*/false, b,
        /*c_mod=*/(short)0, c, /*reuse_a=*/false, /*reuse_b=*/false);
}

__global__ void zero_kernel(float* __restrict__ p, int n) {
    int i = blockIdx.x * blockDim.x + threadIdx.x;
    if (i < n) p[i] = 0.0f;
}

// One block (256 threads = 8 waves) handles 16 edges.
// Wave w computes gate tile [16 edges x 16 cols] (cols n0 = w*16) via WMMA,
// multiplies by message, scatter-adds into the node accumulator.
__global__ void edge_kernel(const float* __restrict__ dist,
                            const float* __restrict__ msg,
                            const int*   __restrict__ tgt,
                            const float* __restrict__ Wd,   // [HIDDEN, NUM_RADIAL]
                            float*       __restrict__ acc,  // [N, HIDDEN]
                            int E) {
    __shared__ float sDist[16 * NUM_RADIAL];  // 16 edges x 16 radial

    const int t    = threadIdx.x;      // 0..255
    const int wave = t >> 5;           // 0..7 -> column tile
    const int lane = t & 31;
    const int e0   = blockIdx.x * 16;

    // Stage dist tile: 256 contiguous floats, one per thread (coalesced).
    {
        const int gidx = e0 * NUM_RADIAL + t;
        sDist[t] = (gidx < E * NUM_RADIAL) ? dist[gidx] : 0.0f;
    }
    __syncthreads();

    const int  n0 = wave * 16;
    const int  lr = lane & 15;
    const bool hi = lane >= 16;

    v8f c = {0.f, 0.f, 0.f, 0.f, 0.f, 0.f, 0.f, 0.f};
    const float* wrow = Wd + (n0 + lr) * NUM_RADIAL;  // column (n0+lr) of Wd^T
    #pragma unroll
    for (int k0 = 0; k0 < NUM_RADIAL; k0 += 4) {
        const int ka = k0 + (hi ? 2 : 0);
        v2f a, b;
        a.x = sDist[lr * NUM_RADIAL + ka];
        a.y = sDist[lr * NUM_RADIAL + ka + 1];
        b.x = wrow[ka];
        b.y = wrow[ka + 1];
        c = wmma_f32_16x16x4(a, b, c);
    }

    // C layout: VGPR r, lanes 0-15 -> (M=r, N=lane); lanes 16-31 -> (M=r+8, N=lane-16)
    const int col   = n0 + lr;
    const int eBase = e0 + (hi ? 8 : 0);

    if (e0 + 16 <= E) {
        // Full-tile fast path: uniform, no per-element EXEC masking.
        // Hoist index + message loads so they can issue as clauses.
        int tg[8];
        #pragma unroll
        for (int r = 0; r < 8; ++r) tg[r] = tgt[eBase + r];
        float mv[8];
        #pragma unroll
        for (int r = 0; r < 8; ++r)
            mv[r] = __builtin_nontemporal_load(&msg[(size_t)(eBase + r) * HIDDEN + col]);
        #pragma unroll
        for (int r = 0; r < 8; ++r)
            atomicAdd(&acc[(size_t)tg[r] * HIDDEN + col], c[r] * mv[r]);
    } else {
        #pragma unroll
        for (int r = 0; r < 8; ++r) {
            const int e = eBase + r;
            if (e < E) {
                const float m = __builtin_nontemporal_load(&msg[(size_t)e * HIDDEN + col]);
                atomicAdd(&acc[(size_t)tgt[e] * HIDDEN + col], c[r] * m);
            }
        }
    }
}

// Y[N,128] = act( X[N,128] @ W[128,128]^T + bias )
// One block per 16-row tile; wave w owns output cols [w*16, w*16+16).
__global__ void mlp_kernel(const float* __restrict__ X,
                           const float* __restrict__ W,     // [HIDDEN, HIDDEN] (out, in)
                           const float* __restrict__ bias,  // [HIDDEN] or nullptr
                           float*       __restrict__ Y,
                           int N, int act) {
    __shared__ float sX[16 * HIDDEN];  // 8 KB

    const int t    = threadIdx.x;
    const int wave = t >> 5;
    const int lane = t & 31;
    const int m0   = blockIdx.x * 16;

    // Coalesced stage of 16 rows (2048 contiguous floats), zero-pad past N.
    for (int i = t; i < 16 * HIDDEN; i += 256) {
        const long g = (long)m0 * HIDDEN + i;
        sX[i] = (g < (long)N * HIDDEN) ? X[g] : 0.0f;
    }
    __syncthreads();

    const int  n0 = wave * 16;
    const int  lr = lane & 15;
    const bool hi = lane >= 16;

    v8f c = {0.f, 0.f, 0.f, 0.f, 0.f, 0.f, 0.f, 0.f};
    const float* wrow = W + (n0 + lr) * HIDDEN;  // B[k][n] = W[n][k]
    #pragma unroll 8
    for (int k0 = 0; k0 < HIDDEN; k0 += 4) {
        const int ka = k0 + (hi ? 2 : 0);
        v2f a, b;
        a.x = sX[lr * HIDDEN + ka];
        a.y = sX[lr * HIDDEN + ka + 1];
        b.x = wrow[ka];
        b.y = wrow[ka + 1];
        c = wmma_f32_16x16x4(a, b, c);
    }

    const int   col = n0 + lr;
    const float bv  = bias ? bias[col] : 0.0f;
    const int   mB  = m0 + (hi ? 8 : 0);

    if (m0 + 16 <= N) {
        // Full-tile fast path: unguarded, clause-friendly stores.
        #pragma unroll
        for (int r = 0; r < 8; ++r) {
            float y = c[r] + bv;
            if (act) y = y / (1.0f + __expf(-y));  // SiLU
            Y[(size_t)(mB + r) * HIDDEN + col] = y;
        }
    } else {
        #pragma unroll
        for (int r = 0; r < 8; ++r) {
            const int m = mB + r;
            if (m < N) {
                float y = c[r] + bv;
                if (act) y = y / (1.0f + __expf(-y));
                Y[(size_t)m * HIDDEN + col] = y;
            }
        }
    }
}

extern "C" void kernel_launch(void* const* d_in, const int* in_sizes, int n_in,
                              void* d_out, int out_size, void* d_ws, size_t ws_size,
                              hipStream_t stream) {
    const float* dist = (const float*)d_in[0];  // [E, 16]
    const float* msg  = (const float*)d_in[1];  // [E, 128]
    const int*   eidx = (const int*)d_in[2];    // [2, E]
    const float* Wd   = (const float*)d_in[4];  // [128, 16]
    const float* Ws   = (const float*)d_in[5];  // [3, 128, 128]
    const float* bs   = (const float*)d_in[6];  // [3, 128]
    const float* Wout = (const float*)d_in[7];  // [128, 128]

    const int E = in_sizes[0] / NUM_RADIAL;
    const int N = out_size / HIDDEN;
    const int* tgt = eidx + E;  // edge_index[1]

    float* acc  = (float*)d_ws;                 // [N, 128]
    float* buf1 = acc + (size_t)N * HIDDEN;     // [N, 128]
    float* out  = (float*)d_out;

    const int nElems = N * HIDDEN;
    zero_kernel<<<(nElems + 255) / 256, 256, 0, stream>>>(acc, nElems);

    edge_kernel<<<(E + 15) / 16, 256, 0, stream>>>(dist, msg, tgt, Wd, acc, E);

    const int mlpBlocks = (N + 15) / 16;
    mlp_kernel<<<mlpBlocks, 256, 0, stream>>>(acc,  Ws,                       bs,              buf1, N, 1);
    mlp_kernel<<<mlpBlocks, 256, 0, stream>>>(buf1, Ws + 1 * HIDDEN * HIDDEN, bs + 1 * HIDDEN, acc,  N, 1);
    mlp_kernel<<<mlpBlocks, 256, 0, stream>>>(acc,  Ws + 2 * HIDDEN * HIDDEN, bs + 2 * HIDDEN, buf1, N, 1);
    mlp_kernel<<<mlpBlocks, 256, 0, stream>>>(buf1, Wout,                     nullptr,         out,  N, 0);
}